// OuterProduct_80092550136303
// MI455X (gfx1250) — compile-verified
//
#include <hip/hip_runtime.h>

// ---------------------------------------------------------------------------
// Problem constants (match reference): B=1, N=2048, NBR=32, C=384, C1=32, OUT=128
// Factorized algorithm:
//   left  = m_i * (s_i @ Wl + bl)                       (2048 x 32)
//   M     = left @ Wo_view(32 x 4096)                   (2048 x 4096)  -> d_ws
//   right = m_j * (s_j @ Wr + br)                       (2048*32 x 32)
//   out_i = right_i(32x32) @ M_i(32x128) + bo           (2048 x 32 x 128)
// Wo_view is Wo (1024x128) reinterpreted row-major as (32 x 4096):
//   (a*32+c)*128+o == a*4096 + (c*128+o).  No data movement needed.
// All matmuls in fp32; all but s_i @ Wl use V_WMMA_F32_16X16X4_F32 (memory-bound workload;
// fp32 WMMA keeps full reference precision).  s_j (the dominant 100 MB
// stream) is staged into double-buffered LDS with CDNA5 async loads
// (GLOBAL_LOAD_ASYNC_TO_LDS_B128, ASYNCcnt) so HBM transfer of chunk k+1
// overlaps WMMA compute of chunk k.
// ---------------------------------------------------------------------------

typedef __attribute__((ext_vector_type(2))) float v2f;
typedef __attribute__((ext_vector_type(4))) float v4f;
typedef __attribute__((ext_vector_type(8))) float v8f;
typedef __attribute__((ext_vector_type(4))) int   v4i;

#define N_RES  2048
#define NBRS   32
#define C_IN   384
#define C1D    32
#define C_OUT  128
#define D_DIM  4096   // C1 * C_OUT  (columns of Wo_view)

#if defined(__AMDGCN__) && defined(__gfx1250__) && \
    __has_builtin(__builtin_amdgcn_global_load_async_to_lds_b128)
#  define HAVE_ASYNC_LDS 1
#else
#  define HAVE_ASYNC_LDS 0
#endif

__device__ __forceinline__ void async_ld_b128(const float* g, float* l) {
#if HAVE_ASYNC_LDS
  // Builtin signature (from hipcc diagnostic): param0 = v4i in AS1 (global),
  // param1 = v4i in AS3 (LDS), then imm offset + cpol.
  __builtin_amdgcn_global_load_async_to_lds_b128(
      (__attribute__((address_space(1))) v4i*)g,
      (__attribute__((address_space(3))) v4i*)l,
      /*imm offset*/ 0, /*cpol*/ 0);
#else
  *(v4f*)l = *(const v4f*)g;
#endif
}

__device__ __forceinline__ void wait_async0() {
#if HAVE_ASYNC_LDS
#  if __has_builtin(__builtin_amdgcn_s_wait_asynccnt)
  __builtin_amdgcn_s_wait_asynccnt(0);
#  else
  asm volatile("s_wait_asynccnt 0x0" ::: "memory");
#  endif
#endif
}

__device__ __forceinline__ v8f wmma4(v2f a, v2f b, v8f c) {
  // 8-arg pattern: (neg_a, A, neg_b, B, c_mod, C, reuse_a, reuse_b)
  return __builtin_amdgcn_wmma_f32_16x16x4_f32(false, a, false, b,
                                               (short)0, c, false, false);
}

// ---------------------------------------------------------------------------
// Kernel 1: one workgroup = 16 residues.
//   left_tile(16x32) = m_i * (s_i_tile @ Wl + bl)   (scalar FMAs from LDS)
//   M_tile(16x4096)  = left_tile @ Wo_view          (WMMA, K=32)
// Grid 128 x 256 threads (8 waves; each wave owns 32 of 256 N-tiles).
// ---------------------------------------------------------------------------
__global__ __launch_bounds__(256) void k_left_M(
    const float* __restrict__ s_i, const float* __restrict__ m_i,
    const float* __restrict__ Wl,  const float* __restrict__ bl,
    const float* __restrict__ Wo,  float* __restrict__ Mbuf)
{
  __shared__ float lds_wl[C_IN * C1D];   // 48 KB
  __shared__ float lds_si[16 * C_IN];    // 24 KB
  __shared__ float lds_left[16 * 33];    // padded stride 33

  const int t  = threadIdx.x;
  const int i0 = blockIdx.x * 16;

  // Stage Wl (12288 floats) and the contiguous s_i tile (6144 floats).
  for (int q = 0; q < 12; ++q) {
    int v = q * 256 + t;
    ((v4f*)lds_wl)[v] = ((const v4f*)Wl)[v];
  }
  const v4f* si_g = (const v4f*)(s_i + (size_t)i0 * C_IN);
  for (int q = 0; q < 6; ++q) {
    int v = q * 256 + t;
    ((v4f*)lds_si)[v] = si_g[v];
  }
  __syncthreads();

  // left projection: 512 outputs, 2 per thread (all lanes of a wave share a row
  // -> broadcast LDS reads of s_i, conflict-free strided reads of Wl).
  for (int q = 0; q < 2; ++q) {
    int idx = q * 256 + t;
    int row = idx >> 5, a = idx & 31;
    float acc = 0.f;
#pragma unroll 8
    for (int e = 0; e < C_IN; ++e)
      acc = fmaf(lds_si[row * C_IN + e], lds_wl[e * C1D + a], acc);
    lds_left[row * 33 + a] = m_i[i0 + row] * (acc + bl[a]);
  }
  __syncthreads();

  const int lane = t & 31, w = t >> 5;
  const int ln = lane & 15;
  const bool hi = lane >= 16;

  // Preload the 8 A-fragments (left_tile is reused by every N-tile).
  v2f afr[8];
#pragma unroll
  for (int kk = 0; kk < 8; ++kk) {
    int k0 = kk * 4 + (hi ? 2 : 0);
    afr[kk].x = lds_left[ln * 33 + k0];
    afr[kk].y = lds_left[ln * 33 + k0 + 1];
  }

  // 256 N-tiles of 16 cols; wave w takes [32w, 32w+32). Wo read once per WG.
  for (int nt = w * 32; nt < w * 32 + 32; ++nt) {
    const int n0 = nt * 16;
    v8f acc = {};
#pragma unroll
    for (int kk = 0; kk < 8; ++kk) {
      int k0 = kk * 4 + (hi ? 2 : 0);
      v2f b;
      b.x = Wo[(size_t)k0 * D_DIM + n0 + ln];
      b.y = Wo[(size_t)(k0 + 1) * D_DIM + n0 + ln];
      acc = wmma4(afr[kk], b, acc);
    }
    float* mrow = Mbuf + (size_t)(i0 + (hi ? 8 : 0)) * D_DIM + n0 + ln;
#pragma unroll
    for (int r = 0; r < 8; ++r)
      mrow[(size_t)r * D_DIM] = acc[r];
  }
}

// ---------------------------------------------------------------------------
// Kernel 2: one workgroup = 8 residues (256 consecutive s_j rows).
//   Phase B: right(256x32) = m_j * (s_j_rows @ Wr + br); s_j streamed through
//            double-buffered LDS via async copy; WMMA K=384 in 12 chunks.
//   Phase C: wave w -> residue i0+w: out_i = right_i(32x32) @ M_i(32x128) + bo.
// Grid 256 x 256 threads.
// LDS: 48 + 2*36 + 35 + 1 KB = 156 KB -> 2 workgroups / WGP.
// ---------------------------------------------------------------------------
__global__ __launch_bounds__(256) void k_right_out(
    const float* __restrict__ s_j, const float* __restrict__ m_j,
    const float* __restrict__ Wr,  const float* __restrict__ br,
    const float* __restrict__ bo,  const float* __restrict__ Mbuf,
    float* __restrict__ out)
{
  __shared__ float lds_wr[C_IN * C1D];     // 48 KB
  __shared__ float lds_sj[2][256 * 36];    // 2 x 36 KB, stride 36 (conflict-free, 16B aligned)
  __shared__ float lds_rt[256 * 35];       // 35 KB, stride 35 (conflict-free A-frag reads)
  __shared__ float lds_mj[256];

  const int t  = threadIdx.x;
  const int i0 = blockIdx.x * 8;
  const int r0 = i0 * NBRS;                // first of 256 consecutive s_j rows

  for (int q = 0; q < 12; ++q) {
    int v = q * 256 + t;
    ((v4f*)lds_wr)[v] = ((const v4f*)Wr)[v];
  }
  lds_mj[t] = m_j[r0 + t];

  const int lane = t & 31, w = t >> 5;
  const int ln = lane & 15;
  const bool hi = lane >= 16;

  // Issue one 256x32 chunk (2048 x b128) into LDS buffer `buf`.
  auto issue_chunk = [&](int kb, int buf) {
#pragma unroll
    for (int q = 0; q < 8; ++q) {
      int v = q * 256 + t;
      int row = v >> 3, c4 = v & 7;        // 8 lanes cover one 128B row segment
      async_ld_b128(s_j + (size_t)(r0 + row) * C_IN + kb * 32 + c4 * 4,
                    &lds_sj[buf][row * 36 + c4 * 4]);
    }
  };

  // Wave w owns m-tiles {2w, 2w+1} x n-tiles {0,1} of the 256x32 right GEMM.
  const int mtb = w * 2;
  v8f acc[2][2] = {{{}, {}}, {{}, {}}};

  issue_chunk(0, 0);
  for (int kb = 0; kb < C_IN / 32; ++kb) {       // 12 K-chunks of 32
    wait_async0();                               // chunk kb landed in LDS
    __syncthreads();                             // ..for every wave; prev buffer free
    if (kb + 1 < C_IN / 32)
      issue_chunk(kb + 1, (kb + 1) & 1);         // overlaps compute below
    const float* sj = lds_sj[kb & 1];
#pragma unroll
    for (int kk = 0; kk < 8; ++kk) {             // 8 WMMA K-steps of 4
      int kloc = kk * 4 + (hi ? 2 : 0);
      v2f a0, a1, b0, b1;
      a0.x = sj[(mtb * 16 + ln) * 36 + kloc];
      a0.y = sj[(mtb * 16 + ln) * 36 + kloc + 1];
      a1.x = sj[((mtb + 1) * 16 + ln) * 36 + kloc];
      a1.y = sj[((mtb + 1) * 16 + ln) * 36 + kloc + 1];
      int kg = kb * 32 + kloc;
      b0.x = lds_wr[kg * C1D + ln];        b0.y = lds_wr[(kg + 1) * C1D + ln];
      b1.x = lds_wr[kg * C1D + 16 + ln];   b1.y = lds_wr[(kg + 1) * C1D + 16 + ln];
      acc[0][0] = wmma4(a0, b0, acc[0][0]);
      acc[0][1] = wmma4(a0, b1, acc[0][1]);
      acc[1][0] = wmma4(a1, b0, acc[1][0]);
      acc[1][1] = wmma4(a1, b1, acc[1][1]);
    }
  }

  // Epilogue: bias + neighbor mask, park right tile in LDS.
#pragma unroll
  for (int mi = 0; mi < 2; ++mi)
#pragma unroll
    for (int nt = 0; nt < 2; ++nt) {
      int acol = nt * 16 + ln;
      float brv = br[acol];
#pragma unroll
      for (int r = 0; r < 8; ++r) {
        int row = (mtb + mi) * 16 + r + (hi ? 8 : 0);
        lds_rt[row * 35 + acol] = (acc[mi][nt][r] + brv) * lds_mj[row];
      }
    }
  __syncthreads();

  // Phase C: wave w handles residue ig = i0+w.
  // out_i(32x128) = right_i(32x32) @ M_i(32x128): 2 m-tiles x 8 n-tiles, K=32.
  const int ig = i0 + w;
  v2f afr[2][8];
#pragma unroll
  for (int mt = 0; mt < 2; ++mt)
#pragma unroll
    for (int kk = 0; kk < 8; ++kk) {
      int k0 = kk * 4 + (hi ? 2 : 0);
      afr[mt][kk].x = lds_rt[(w * 32 + mt * 16 + ln) * 35 + k0];
      afr[mt][kk].y = lds_rt[(w * 32 + mt * 16 + ln) * 35 + k0 + 1];
    }
  const float* Mi = Mbuf + (size_t)ig * D_DIM;   // L2-resident
#pragma unroll
  for (int nt = 0; nt < 8; ++nt) {
    const int n0 = nt * 16;
    v8f o0 = {}, o1 = {};
#pragma unroll
    for (int kk = 0; kk < 8; ++kk) {
      int k0 = kk * 4 + (hi ? 2 : 0);
      v2f b;                                      // shared by both m-tiles
      b.x = Mi[(size_t)k0 * C_OUT + n0 + ln];
      b.y = Mi[(size_t)(k0 + 1) * C_OUT + n0 + ln];
      o0 = wmma4(afr[0][kk], b, o0);
      o1 = wmma4(afr[1][kk], b, o1);
    }
    float bov = bo[n0 + ln];
    float* obase = out + ((size_t)ig * NBRS + (hi ? 8 : 0)) * C_OUT + n0 + ln;
#pragma unroll
    for (int r = 0; r < 8; ++r) {
      obase[(size_t)r * C_OUT]        = o0[r] + bov;   // k = r (+8 if hi)
      obase[(size_t)(16 + r) * C_OUT] = o1[r] + bov;   // k = 16 + r (+8 if hi)
    }
  }
}

// ---------------------------------------------------------------------------
// Inputs (setup_inputs order): 0 s_i, 1 s_j, 2 m_i, 3 m_j, 4 Wl, 5 bl,
//                              6 Wr, 7 br, 8 Wo, 9 bo.  All float32.
// d_ws holds M (2048 x 4096 fp32 = 32 MiB).
// ---------------------------------------------------------------------------
extern "C" void kernel_launch(void* const* d_in, const int* in_sizes, int n_in,
                              void* d_out, int out_size, void* d_ws, size_t ws_size,
                              hipStream_t stream) {
  const float* s_i = (const float*)d_in[0];
  const float* s_j = (const float*)d_in[1];
  const float* m_i = (const float*)d_in[2];
  const float* m_j = (const float*)d_in[3];
  const float* Wl  = (const float*)d_in[4];
  const float* bl  = (const float*)d_in[5];
  const float* Wr  = (const float*)d_in[6];
  const float* br  = (const float*)d_in[7];
  const float* Wo  = (const float*)d_in[8];
  const float* bo  = (const float*)d_in[9];
  float* outp = (float*)d_out;
  float* Mbuf = (float*)d_ws;   // requires N_RES * D_DIM * 4 = 32 MiB scratch

  k_left_M<<<N_RES / 16, 256, 0, stream>>>(s_i, m_i, Wl, bl, Wo, Mbuf);
  k_right_out<<<N_RES / 8, 256, 0, stream>>>(s_j, m_j, Wr, br, bo, Mbuf, outp);
}